// IrrelevanceFilter_773094113753
// MI455X (gfx1250) — compile-verified
//
#include <hip/hip_runtime.h>
#include <hip/hip_bf16.h>
#include <math.h>

#define B_  32
#define T_  4096
#define D_  256
#define H_  128
#define BT_ (B_*T_)
#define TCHUNK_ 32           // k_topic partial chunks per batch

typedef __attribute__((ext_vector_type(2))) float v2f;
typedef __attribute__((ext_vector_type(8))) float v8f;

// ---------------------------------------------------------------------------
// 1) attention scores: w[b,t] = dot(q, x[b,t,:]) / 0.5
//    One wave per token: lanes read consecutive float4 -> fully coalesced.
// ---------------------------------------------------------------------------
__global__ __launch_bounds__(256) void k_scores(const float* __restrict__ x,
                                                const float* __restrict__ tq,
                                                float* __restrict__ w)
{
    __shared__ float qs[D_];
    int tid = threadIdx.x;
    qs[tid] = tq[tid];
    __syncthreads();
    int wave = tid >> 5, lane = tid & 31;
    size_t tok = (size_t)blockIdx.x * 8 + wave;
    const float4* xr = (const float4*)(x + tok * D_);
    const float4* qv = (const float4*)qs;
    float acc = 0.f;
#pragma unroll
    for (int i = 0; i < 2; ++i) {              // 2 * 32 lanes * 4 = 256 elems
        float4 a  = xr[i * 32 + lane];
        float4 q4 = qv[i * 32 + lane];
        acc += a.x * q4.x + a.y * q4.y + a.z * q4.z + a.w * q4.w;
    }
    for (int off = 1; off < 32; off <<= 1) acc += __shfl_xor(acc, off, 32);
    if (lane == 0) w[tok] = acc * 2.0f;        // 1 / TEMPERATURE
}

// ---------------------------------------------------------------------------
// 2) softmax over T per batch (in place on w)
// ---------------------------------------------------------------------------
__global__ __launch_bounds__(256) void k_softmax(float* __restrict__ w)
{
    __shared__ float red[256];
    int tid = threadIdx.x;
    float* wb = w + (size_t)blockIdx.x * T_;

    float m = -1e30f;
    for (int j = tid; j < T_; j += 256) m = fmaxf(m, wb[j]);
    red[tid] = m; __syncthreads();
    for (int s = 128; s > 0; s >>= 1) {
        if (tid < s) red[tid] = fmaxf(red[tid], red[tid + s]);
        __syncthreads();
    }
    float M = red[0]; __syncthreads();

    float sum = 0.f;
    for (int j = tid; j < T_; j += 256) {
        float e = __expf(wb[j] - M);
        wb[j] = e; sum += e;
    }
    red[tid] = sum; __syncthreads();
    for (int s = 128; s > 0; s >>= 1) {
        if (tid < s) red[tid] += red[tid + s];
        __syncthreads();
    }
    float inv = 1.0f / red[0];
    for (int j = tid; j < T_; j += 256) wb[j] *= inv;
}

// ---------------------------------------------------------------------------
// 3) main_topic[b,d] = sum_t w[b,t] * x[b,t,d]  (two-stage for parallelism:
//    1024 partial blocks stream x at full bandwidth, then a tiny reduce)
// ---------------------------------------------------------------------------
__global__ __launch_bounds__(256) void k_topic_partial(const float* __restrict__ x,
                                                       const float* __restrict__ w,
                                                       float* __restrict__ part)
{
    const int CH = T_ / TCHUNK_;               // 128 timesteps per chunk
    int d = threadIdx.x;
    int b = blockIdx.x;
    int c = blockIdx.y;
    const float* xb = x + ((size_t)b * T_ + (size_t)c * CH) * D_;
    const float* wb = w + (size_t)b * T_ + (size_t)c * CH;
    float acc = 0.f;
    for (int t = 0; t < CH; ++t)
        acc += wb[t] * xb[(size_t)t * D_ + d];
    part[((size_t)b * TCHUNK_ + c) * D_ + d] = acc;
}

__global__ __launch_bounds__(256) void k_topic_reduce(const float* __restrict__ part,
                                                      float* __restrict__ mt)
{
    int d = threadIdx.x;
    int b = blockIdx.x;
    float acc = 0.f;
#pragma unroll 4
    for (int c = 0; c < TCHUNK_; ++c)
        acc += part[((size_t)b * TCHUNK_ + c) * D_ + d];
    mt[b * D_ + d] = acc;
}

// ---------------------------------------------------------------------------
// 4) encoder: Linear(256x256)+LN+GELU+Linear(256x128), f32 WMMA 16x16x4
//    mode 0: write l2-normalized encoding to out_tenc (topic rows)
//    mode 1: write cosine(relevance) vs tenc to out_rel
//    block = 128 threads = 4 waves; wave handles 16 tokens.
//    LDS: h stored TRANSPOSED [k][m] per wave -> conflict-free LN & A-frags.
// ---------------------------------------------------------------------------
__global__ __launch_bounds__(128) void k_encode(
    const float* __restrict__ X, int ntok,
    const float* __restrict__ W1, const float* __restrict__ b1,
    const float* __restrict__ g,  const float* __restrict__ be,
    const float* __restrict__ W2, const float* __restrict__ b2,
    const float* __restrict__ tenc,
    float* __restrict__ out_tenc,
    float* __restrict__ out_rel,
    int mode)
{
    extern __shared__ float hb[];            // [4 waves][256 k][16 m] = 64KB
    int tid   = threadIdx.x;
    int wave  = tid >> 5;
    int lane  = tid & 31;
    int halve = lane >> 4;                   // 0 = lanes 0-15, 1 = lanes 16-31
    int mrow  = lane & 15;                   // M row / N col inside tile
    int rowBase = blockIdx.x * 64 + wave * 16;
    int row  = rowBase + mrow;
    int rowC = row < ntok ? row : (ntok - 1);
    const float* Arow = X + (size_t)rowC * D_;
    float* hw = hb + wave * (D_ * 16);

    // ---- GEMM1: h = X @ W1 + b1 ; store transposed into LDS ----
    for (int nt = 0; nt < 16; ++nt) {
        int ncol = nt * 16 + mrow;
        v8f acc = {};
        for (int ks = 0; ks < 64; ++ks) {
            int k0 = ks * 4 + halve * 2;     // f32 A-layout: K = (lane/16)*2 + v
            v2f a;  a.x  = Arow[k0];
                    a.y  = Arow[k0 + 1];
            v2f bb; bb.x = W1[(size_t)k0       * D_ + ncol];
                    bb.y = W1[(size_t)(k0 + 1) * D_ + ncol];
            acc = __builtin_amdgcn_wmma_f32_16x16x4_f32(
                      false, a, false, bb, (short)0, acc, false, false);
        }
        float bias = b1[ncol];
#pragma unroll
        for (int r = 0; r < 8; ++r)          // C: vgpr r -> M = r + halve*8
            hw[ncol * 16 + (r + halve * 8)] = acc[r] + bias;
    }
    __syncthreads();

    // ---- LayerNorm + exact GELU (per token row, 2 lanes per row) ----
    {
        float sum = 0.f, sq = 0.f;
        int i0 = halve * 128;
        for (int k = i0; k < i0 + 128; ++k) {
            float v = hw[k * 16 + mrow];
            sum += v; sq += v * v;
        }
        sum += __shfl_xor(sum, 16, 32);
        sq  += __shfl_xor(sq,  16, 32);
        float mu   = sum * (1.0f / 256.0f);
        float var  = sq  * (1.0f / 256.0f) - mu * mu;
        float rstd = rsqrtf(var + 1e-5f);
        for (int k = i0; k < i0 + 128; ++k) {
            float v = (hw[k * 16 + mrow] - mu) * rstd * g[k] + be[k];
            v = 0.5f * v * (1.0f + erff(v * 0.70710678118654752f));
            hw[k * 16 + mrow] = v;
        }
    }
    __syncthreads();

    // ---- GEMM2: e = h @ W2 + b2 ; keep all 8 tiles in registers ----
    v8f acc2[8];
    for (int nt = 0; nt < 8; ++nt) {
        int ncol = nt * 16 + mrow;
        v8f acc = {};
        for (int ks = 0; ks < 64; ++ks) {
            int k0 = ks * 4 + halve * 2;
            v2f a;  a.x  = hw[k0       * 16 + mrow];
                    a.y  = hw[(k0 + 1) * 16 + mrow];
            v2f bb; bb.x = W2[(size_t)k0       * H_ + ncol];
                    bb.y = W2[(size_t)(k0 + 1) * H_ + ncol];
            acc = __builtin_amdgcn_wmma_f32_16x16x4_f32(
                      false, a, false, bb, (short)0, acc, false, false);
        }
        float bias = b2[ncol];
#pragma unroll
        for (int r = 0; r < 8; ++r) acc[r] += bias;
        acc2[nt] = acc;
    }

    // ---- per-row L2 norm + output (reductions inside each wave half) ----
    float tv[8];
    if (mode == 1) {
        int bIdx = (blockIdx.x * 64) / T_;   // block never straddles a batch
#pragma unroll
        for (int nt = 0; nt < 8; ++nt)
            tv[nt] = tenc[(size_t)bIdx * H_ + nt * 16 + mrow];
    }
#pragma unroll
    for (int r = 0; r < 8; ++r) {
        float s = 0.f;
#pragma unroll
        for (int nt = 0; nt < 8; ++nt) s += acc2[nt][r] * acc2[nt][r];
        for (int off = 1; off < 16; off <<= 1) s += __shfl_xor(s, off, 32);
        float inv = 1.0f / fmaxf(sqrtf(s), 1e-12f);
        int rr = rowBase + halve * 8 + r;
        if (mode == 0) {
            if (rr < ntok) {
#pragma unroll
                for (int nt = 0; nt < 8; ++nt)
                    out_tenc[(size_t)rr * H_ + nt * 16 + mrow] = acc2[nt][r] * inv;
            }
        } else {
            float d = 0.f;
#pragma unroll
            for (int nt = 0; nt < 8; ++nt) d += acc2[nt][r] * tv[nt];
            for (int off = 1; off < 16; off <<= 1) d += __shfl_xor(d, off, 32);
            d *= inv;
            if (mrow == r && rr < ntok) out_rel[rr] = d;
        }
    }
}

// ---------------------------------------------------------------------------
// 5) transpose x[B,T,D] -> xt[B,D,T] (32x32 LDS tiles)
// ---------------------------------------------------------------------------
__global__ __launch_bounds__(256) void k_transpose_fwd(const float* __restrict__ x,
                                                       float* __restrict__ xt)
{
    __shared__ float tile[32][33];
    int b  = blockIdx.z;
    int t0 = blockIdx.x * 32, d0 = blockIdx.y * 32;
    int tx = threadIdx.x, ty = threadIdx.y;            // 32 x 8
    for (int i = ty; i < 32; i += 8)
        tile[i][tx] = x[((size_t)b * T_ + t0 + i) * D_ + d0 + tx];
    __syncthreads();
    for (int i = ty; i < 32; i += 8)
        xt[((size_t)b * D_ + d0 + i) * T_ + t0 + tx] = tile[tx][i];
}

__global__ __launch_bounds__(256) void k_transpose_bwd(const float* __restrict__ xt,
                                                       float* __restrict__ y)
{
    __shared__ float tile[32][33];
    int b  = blockIdx.z;
    int d0 = blockIdx.x * 32, t0 = blockIdx.y * 32;
    int tx = threadIdx.x, ty = threadIdx.y;
    for (int i = ty; i < 32; i += 8)
        tile[i][tx] = xt[((size_t)b * D_ + d0 + i) * T_ + t0 + tx];
    __syncthreads();
    for (int i = ty; i < 32; i += 8)
        y[((size_t)b * T_ + t0 + i) * D_ + d0 + tx] = tile[tx][i];
}

// ---------------------------------------------------------------------------
// 6) fft(4096) -> masked phase filter -> ifft, entirely in LDS.
//    Stockham radix-2 ping-pong (no bit reversal). 12 stages (even) -> result
//    lands back in the (ar,ai) arrays. Row is brought in through the CDNA5
//    async global->LDS path (ASYNCcnt), no VGPR round-trip.
// ---------------------------------------------------------------------------
__device__ __forceinline__ void fft4096(float* sr, float* si,
                                        float* dr, float* di,
                                        float dsign, int tid)
{
    for (int Ns = 1; Ns < 4096; Ns <<= 1) {
        for (int j = tid; j < 2048; j += 256) {
            int   p  = j & (Ns - 1);
            float ur = sr[j],        ui = si[j];
            float vr = sr[j + 2048], vi = si[j + 2048];
            float ang = dsign * 6.28318530717958647f * (float)p / (float)(Ns << 1);
            float s, c;
            __sincosf(ang, &s, &c);
            float tr = vr * c - vi * s;
            float ti = vr * s + vi * c;
            int idx = ((j - p) << 1) + p;
            dr[idx]      = ur + tr;  di[idx]      = ui + ti;
            dr[idx + Ns] = ur - tr;  di[idx + Ns] = ui - ti;
        }
        __syncthreads();
        float* t;
        t = sr; sr = dr; dr = t;
        t = si; si = di; di = t;
    }
}

__global__ __launch_bounds__(256) void k_fft(float* __restrict__ xt,
                                             const float* __restrict__ rel,
                                             const float* __restrict__ sf)
{
    extern __shared__ float smem[];
    float* ar = smem;
    float* ai = smem + 4096;
    float* br = smem + 8192;
    float* bi = smem + 12288;
    int tid = threadIdx.x;
    int idx = blockIdx.x;                    // [0, B*D)
    int b   = idx / D_;
    int d   = idx - b * D_;
    float* rowp = xt + (size_t)idx * T_;

    // async global -> LDS copy of the 16KB row (4 x b128 per lane set);
    // LDS aperture maps generic addr[31:0] to the LDS offset.
#pragma unroll
    for (int it = 0; it < 4; ++it) {
        int j = it * 1024 + tid * 4;
        unsigned int lds_off = (unsigned int)(uintptr_t)(ar + j);
        const float* gp = rowp + j;
        asm volatile("global_load_async_to_lds_b128 %0, %1, off"
                     :: "v"(lds_off), "v"(gp) : "memory");
    }
    for (int j = tid; j < T_; j += 256) ai[j] = 0.f;
    asm volatile("s_wait_asynccnt 0x0" ::: "memory");
    __syncthreads();

    fft4096(ar, ai, br, bi, -1.0f, tid);     // forward

    float phi = sf[d & 3];                   // quaternion component = d % 4
    float sn, cs;
    __sincosf(phi, &sn, &cs);
    const float* rb = rel + (size_t)b * T_;
    for (int j = tid; j < T_; j += 256) {
        if (rb[j] < 0.4f) {                  // irrelevant -> rotate by e^{i phi}
            float re = ar[j], im = ai[j];
            ar[j] = re * cs - im * sn;
            ai[j] = re * sn + im * cs;
        }
    }
    __syncthreads();

    fft4096(ar, ai, br, bi, 1.0f, tid);      // inverse (scale below)

    const float invN = 1.0f / 4096.0f;
    for (int j = tid; j < T_; j += 256) rowp[j] = ar[j] * invN;   // real part
}

// ---------------------------------------------------------------------------
extern "C" void kernel_launch(void* const* d_in, const int* in_sizes, int n_in,
                              void* d_out, int out_size, void* d_ws, size_t ws_size,
                              hipStream_t stream)
{
    (void)in_sizes; (void)n_in; (void)out_size; (void)ws_size;
    const float* x  = (const float*)d_in[0];
    const float* W1 = (const float*)d_in[1];
    const float* b1 = (const float*)d_in[2];
    const float* lg = (const float*)d_in[3];
    const float* lb = (const float*)d_in[4];
    const float* W2 = (const float*)d_in[5];
    const float* b2 = (const float*)d_in[6];
    const float* tq = (const float*)d_in[7];
    const float* sf = (const float*)d_in[8];
    float* out = (float*)d_out;

    // workspace layout
    float* wsf  = (float*)d_ws;
    float* w    = wsf;                       // [B*T] softmax weights
    float* rel  = w + BT_;                   // [B*T] relevance
    float* mt   = rel + BT_;                 // [B*D] main topic
    float* ten  = mt + B_ * D_;              // [B*H] normalized topic enc
    float* part = ten + B_ * H_;             // [B*TCHUNK*D] topic partials
    uintptr_t p = (uintptr_t)(part + (size_t)B_ * TCHUNK_ * D_);
    p = (p + 255) & ~(uintptr_t)255;
    float* xt = (float*)p;                   // [B*D*T] transposed plane (134MB)

    k_scores <<<BT_ / 8, 256, 0, stream>>>(x, tq, w);
    k_softmax<<<B_,      256, 0, stream>>>(w);
    k_topic_partial<<<dim3(B_, TCHUNK_), 256, 0, stream>>>(x, w, part);
    k_topic_reduce <<<B_,                256, 0, stream>>>(part, mt);

    size_t enc_sh = (size_t)(4 * D_ * 16) * sizeof(float);   // 64KB
    k_encode<<<1,        128, enc_sh, stream>>>(mt, B_,  W1, b1, lg, lb, W2, b2,
                                                nullptr, ten, nullptr, 0);
    k_encode<<<BT_ / 64, 128, enc_sh, stream>>>(x,  BT_, W1, b1, lg, lb, W2, b2,
                                                ten, nullptr, rel, 1);

    dim3 tb(32, 8);
    k_transpose_fwd<<<dim3(T_ / 32, D_ / 32, B_), tb, 0, stream>>>(x, xt);
    k_fft<<<B_ * D_, 256, 64 * 1024, stream>>>(xt, rel, sf);
    k_transpose_bwd<<<dim3(D_ / 32, T_ / 32, B_), tb, 0, stream>>>(xt, out);
}